// RuleNetwork_51419348467936
// MI455X (gfx1250) — compile-verified
//
#include <hip/hip_runtime.h>
#include <math.h>

typedef __attribute__((ext_vector_type(2))) float v2f;
typedef __attribute__((ext_vector_type(8))) float v8f;
typedef __attribute__((ext_vector_type(4))) unsigned int v4u;
typedef __attribute__((ext_vector_type(8))) int v8i;
typedef __attribute__((ext_vector_type(4))) int v4i;

#define V_      8
#define H_      512
#define R_      16
#define RDIM_   64
#define DK_     32
#define GH_     16
#define OUT_    512
#define HPAD_   516                 // 512 + 2 pads x 2 (TDM inserts 2 dwords per 256-dword interval)
#define TEMPR_  5.65685424949238f   // sqrt(d_k) = sqrt(32)
#define EPSG_   1e-10f

__device__ __forceinline__ float gumbel_noise(float u) {
    return -logf(-logf(u + EPSG_) + EPSG_);
}

// hid row layout (TDM-padded): col c lives at c + 2*(c>=256)
__device__ __forceinline__ float hget(const float* hrow, int c) {
    return hrow[c + ((c >> 8) << 1)];
}

// One block = 128 threads (4 waves) handles TWO batch elements:
//   b0 -> WMMA M rows 0..7, b1 -> M rows 8..15 (full 16x16 tile utilization).
// wave = (matsel: q/k) x (N-tile: cols 0-15 / 16-31); K=512 swept in 4-wide steps.
extern "C" __global__ __launch_bounds__(128)
void rule_network_fused(const float* __restrict__ hidden,
                        const float* __restrict__ u_var,
                        const float* __restrict__ u_rule,
                        const float* __restrict__ Wq, const float* __restrict__ bq,
                        const float* __restrict__ Wk, const float* __restrict__ bk,
                        const float* __restrict__ rule_emb,
                        const float* __restrict__ Wq2, const float* __restrict__ bq2,
                        const float* __restrict__ Wk2, const float* __restrict__ bk2,
                        const float* __restrict__ W1, const float* __restrict__ W2,
                        float* __restrict__ out)
{
    __shared__ __align__(16) float hid[16 * HPAD_]; // 2 batches x 8 rows, pitch 516
    __shared__ float qs[2 * 16 * DK_];              // [q|k][16 rows][32]
    __shared__ float hmean[2 * H_];
    __shared__ __align__(16) float routs[2 * OUT_];
    __shared__ float k2s[R_ * DK_];
    __shared__ float vals[2 * 64];
    __shared__ float rvals[2 * R_];
    __shared__ float q2s[2 * DK_];
    __shared__ float h1s[2 * GH_];
    __shared__ int   sel[2 * 3];                    // per sub-batch: istar, jstar, rstar

    const int tid = threadIdx.x;
    const int b0  = blockIdx.x * 2;
    const float* hbase = hidden + (size_t)b0 * (V_ * H_);

    // ---- Tensor Data Mover: async DMA hidden[b0..b1] (16 rows x 512 f32) into LDS,
    //      with 2-dword padding inserted every 256 dwords -> row pitch 516. Wave 0 issues.
    if (tid == 0) {
        __builtin_prefetch(Wq, 0, 3);
        __builtin_prefetch(Wk, 0, 3);
        __builtin_prefetch(W1, 0, 3);
        __builtin_prefetch(W2, 0, 3);

        unsigned lds_base = (unsigned)(size_t)(void*)&hid[0];
        unsigned long long ga = (unsigned long long)(size_t)(const void*)hbase;
        v4u g0;
        g0.x = 1u;                                        // count=1, user descriptor
        g0.y = lds_base;                                  // lds_addr (bytes)
        g0.z = (unsigned)(ga & 0xffffffffu);              // global_addr[31:0]
        g0.w = (unsigned)((ga >> 32) & 0x01ffffffu)       // global_addr[56:32]
             | (2u << 30);                                // type = 2 (image)
        v8i g1;
        unsigned d0 = 0;
        d0 |= (2u << 16);                                 // data_size = 4 bytes
        d0 |= (1u << 20);                                 // pad_enable
        d0 |= (7u << 22);                                 // pad_interval = 256 dwords
        d0 |= (1u << 25);                                 // pad_amount   = 2 dwords
        g1[0] = (int)d0;
        const unsigned td0 = 512u, td1 = 16u;             // tensor dims (elements)
        g1[1] = (int)((td0 & 0xffffu) << 16);             // [63:48] = tensor_dim0 lo16
        g1[2] = (int)(((td0 >> 16) & 0xffffu) | ((td1 & 0xffffu) << 16));
        g1[3] = (int)(((td1 >> 16) & 0xffffu) | (512u << 16));  // tile_dim0 = 512
        g1[4] = (int)(16u);                               // tile_dim1 = 16, tile_dim2 = 0
        g1[5] = (int)512u;                                // tensor_dim0_stride lo32
        g1[6] = 0;                                        // stride hi16 | dim1_stride lo16
        g1[7] = 0;
        v4i g2 = {0, 0, 0, 0};
        v4i g3 = {0, 0, 0, 0};
        v8i g4 = {0, 0, 0, 0, 0, 0, 0, 0};                // unused trailing group (6-arg form)
        __builtin_amdgcn_tensor_load_to_lds(g0, g1, g2, g3, g4, 0);
    }

    // ---- overlap the DMA: rule keys k2 = rule_emb @ Wk2 + bk2 (batch-invariant) ----
    for (int o = tid; o < R_ * DK_; o += 128) {
        int r = o >> 5, n = o & 31;
        float s = bk2[n];
        for (int d = 0; d < RDIM_; ++d) s += rule_emb[r * RDIM_ + d] * Wk2[d * DK_ + n];
        k2s[o] = s;
    }
    if (tid == 0) __builtin_amdgcn_s_wait_tensorcnt(0);
    __syncthreads();

    // ---- q/k projections via V_WMMA_F32_16X16X4_F32 ----
    const int lane   = tid & 31;
    const int wave   = tid >> 5;
    const int matsel = wave >> 1;            // 0 = q, 1 = k
    const int n0     = (wave & 1) * 16;      // N-tile base
    const int mrow   = lane & 15;            // A-matrix row (M)
    const int khalf  = lane >> 4;            // K pair select
    const int nc     = lane & 15;            // B/D column within tile

    const float* Wsel = matsel ? Wk : Wq;
    const float* bsel = matsel ? bk : bq;
    const float* arow = &hid[mrow * HPAD_];

    v8f acc = {0.f, 0.f, 0.f, 0.f, 0.f, 0.f, 0.f, 0.f};
    for (int kk = 0; kk < H_; kk += 4) {
        int c = kk + 2 * khalf;
        int ca = c + ((kk >> 8) << 1);       // TDM pad fixup (uniform per kk group)
        v2f a, bf;
        a.x  = arow[ca];
        a.y  = arow[ca + 1];
        bf.x = Wsel[c * DK_ + n0 + nc];
        bf.y = Wsel[(c + 1) * DK_ + n0 + nc];
        acc = __builtin_amdgcn_wmma_f32_16x16x4_f32(
            false, a, false, bf, (short)0, acc, false, false);
    }
    {
        float bias = bsel[n0 + nc];
        #pragma unroll
        for (int r = 0; r < 8; ++r) {
            int M = r + 8 * khalf;           // rows 0-7: b0 ; rows 8-15: b1
            qs[matsel * (16 * DK_) + M * DK_ + n0 + nc] = acc[r] + bias;
        }
    }
    __syncthreads();

    // ---- 8x8 score grid + Gumbel, argmax -> (i*, j*) per sub-batch ----
    {
        int sub = tid >> 6, tt = tid & 63;
        int i = tt >> 3, j = tt & 7;
        const float* qrow = &qs[(sub * 8 + i) * DK_];
        const float* krow = &qs[16 * DK_ + (sub * 8 + j) * DK_];
        float s = 0.f;
        #pragma unroll
        for (int d = 0; d < DK_; ++d) s += qrow[d] * krow[d];
        float u = u_var[(size_t)(b0 + sub) * 64 + tt];
        vals[sub * 64 + tt] = s / TEMPR_ + gumbel_noise(u);
    }
    __syncthreads();
    if (tid < 2) {
        const float* v = &vals[tid * 64];
        int best = 0; float bv = v[0];
        for (int t = 1; t < 64; ++t) if (v[t] > bv) { bv = v[t]; best = t; }
        sel[tid * 3 + 0] = best >> 3;        // i* : context row
        sel[tid * 3 + 1] = best & 7;         // j* : primary row
    }
    // hmean (independent of selection)
    for (int idx = tid; idx < 2 * H_; idx += 128) {
        int sub = idx >> 9, c = idx & (H_ - 1);
        float s = 0.f;
        #pragma unroll
        for (int vv = 0; vv < V_; ++vv) s += hget(&hid[(sub * 8 + vv) * HPAD_], c);
        hmean[idx] = s * (1.0f / V_);
    }
    __syncthreads();

    // ---- q2 = [h_p | h_c | h_mean] @ Wq2 + bq2 ----
    if (tid < 64) {
        int sub = tid >> 5, n = tid & 31;
        int istar = sel[sub * 3 + 0], jstar = sel[sub * 3 + 1];
        const float* hp = &hid[(sub * 8 + jstar) * HPAD_];
        const float* hc = &hid[(sub * 8 + istar) * HPAD_];
        const float* hm = &hmean[sub * H_];
        float s = bq2[n];
        for (int d = 0; d < H_; ++d) s += hget(hp, d) * Wq2[d * DK_ + n];
        for (int d = 0; d < H_; ++d) s += hget(hc, d) * Wq2[(H_ + d) * DK_ + n];
        for (int d = 0; d < H_; ++d) s += hm[d] * Wq2[(2 * H_ + d) * DK_ + n];
        q2s[tid] = s;
    }
    __syncthreads();

    // ---- rule scores + Gumbel, argmax -> r* ----
    if (tid < 32) {
        int sub = tid >> 4, r = tid & 15;
        float s = 0.f;
        #pragma unroll
        for (int n = 0; n < DK_; ++n) s += q2s[sub * DK_ + n] * k2s[r * DK_ + n];
        float u = u_rule[(size_t)(b0 + sub) * R_ + r];
        rvals[tid] = s / TEMPR_ + gumbel_noise(u);
    }
    __syncthreads();
    if (tid < 2) {
        const float* v = &rvals[tid * R_];
        int best = 0; float bv = v[0];
        for (int r = 1; r < R_; ++r) if (v[r] > bv) { bv = v[r]; best = r; }
        sel[tid * 3 + 2] = best;
    }
    __syncthreads();

    // ---- selected-rule MLP: h1 = relu([h_p|h_c] @ W1[r*]) ----
    if (tid < 32) {
        int sub = tid >> 4, e = tid & 15;
        int istar = sel[sub * 3 + 0], jstar = sel[sub * 3 + 1], rstar = sel[sub * 3 + 2];
        const float* hp = &hid[(sub * 8 + jstar) * HPAD_];
        const float* hc = &hid[(sub * 8 + istar) * HPAD_];
        const float* w1 = W1 + (size_t)rstar * (2 * H_) * GH_;
        float s = 0.f;
        for (int d = 0; d < H_; ++d) s += hget(hp, d) * w1[d * GH_ + e];
        for (int d = 0; d < H_; ++d) s += hget(hc, d) * w1[(H_ + d) * GH_ + e];
        h1s[tid] = s > 0.f ? s : 0.f;
    }
    __syncthreads();

    // ---- rout = h1 @ W2[r*] ----
    for (int idx = tid; idx < 2 * OUT_; idx += 128) {
        int sub = idx >> 9, o = idx & (OUT_ - 1);
        int rstar = sel[sub * 3 + 2];
        const float* w2 = W2 + (size_t)rstar * GH_ * OUT_;
        float s = 0.f;
        #pragma unroll
        for (int e = 0; e < GH_; ++e) s += h1s[sub * GH_ + e] * w2[e * OUT_ + o];
        routs[idx] = s;
    }
    __syncthreads();

    // ---- scatter (vectorized b128 stores): out[b, v, :] = (v == j*) ? rout : 0 ----
    float4* obase4 = (float4*)(out + (size_t)b0 * (V_ * OUT_));
    const float4* r4 = (const float4*)routs;
    const float4 z4 = make_float4(0.f, 0.f, 0.f, 0.f);
    for (int c = tid; c < 2 * V_ * OUT_ / 4; c += 128) {   // 2048 float4 chunks
        int sub = c >> 10, rem = c & 1023;
        int vv = rem >> 7, o4 = rem & 127;
        int jstar = sel[sub * 3 + 1];
        obase4[c] = (vv == jstar) ? r4[sub * (OUT_ / 4) + o4] : z4;
    }
}

extern "C" void kernel_launch(void* const* d_in, const int* in_sizes, int n_in,
                              void* d_out, int out_size, void* d_ws, size_t ws_size,
                              hipStream_t stream) {
    const float* hidden   = (const float*)d_in[0];
    const float* u_var    = (const float*)d_in[1];
    const float* u_rule   = (const float*)d_in[2];
    const float* Wq       = (const float*)d_in[3];
    const float* bq       = (const float*)d_in[4];
    const float* Wk       = (const float*)d_in[5];
    const float* bk       = (const float*)d_in[6];
    const float* rule_emb = (const float*)d_in[7];
    const float* Wq2      = (const float*)d_in[8];
    const float* bq2      = (const float*)d_in[9];
    const float* Wk2      = (const float*)d_in[10];
    const float* bk2      = (const float*)d_in[11];
    const float* W1       = (const float*)d_in[12];
    const float* W2       = (const float*)d_in[13];
    float* out            = (float*)d_out;

    const int B = in_sizes[0] / (V_ * H_);   // 4096
    dim3 grid(B / 2);                        // 2 batch elements per block
    dim3 block(128);
    rule_network_fused<<<grid, block, 0, stream>>>(
        hidden, u_var, u_rule, Wq, bq, Wk, bk, rule_emb,
        Wq2, bq2, Wk2, bk2, W1, W2, out);
}